// CurvatureLoss_2757369004563
// MI455X (gfx1250) — compile-verified
//
#include <hip/hip_runtime.h>
#include <hip/hip_bf16.h>
#include <math.h>

// ---------------------------------------------------------------------------
// Curvature loss, 32 x 1 x 1024 x 1024 f32 -> scalar.
// Memory-bound stencil: LDS-tiled, async global->LDS copies (gfx1250 path),
// wave32 shuffle reduction, two-stage scalar reduce (no atomic hotspot).
// ---------------------------------------------------------------------------

#define EPS_F 1e-8f
static constexpr float VOXW  = 0.061f;
static constexpr int   W     = 1024;        // row width
static constexpr int   ROWS  = 14;          // interior rows per block (73*14 == 1022)
static constexpr int   LROWS = ROWS + 2;    // +halo
static constexpr int   RBLK  = 73;          // row-blocks per image
static constexpr int   NIMG  = 32;
static constexpr int   NTHREADS = 256;      // 8 waves (wave32)

// ---- gfx1250 async global->LDS detection ----------------------------------
#if defined(__has_builtin)
#  if __has_builtin(__builtin_amdgcn_global_load_async_to_lds_b128)
#    define ASYNC_LDS 1
#  endif
#  if __has_builtin(__builtin_amdgcn_s_wait_asynccnt)
#    define WAIT_ASYNC_BUILTIN 1
#  endif
#endif

#if defined(ASYNC_LDS)
// Builtin signature (from clang diagnostic): param0 is
//   'int __vector(4) __device__ *'  i.e. v4i in global addrspace(1)
typedef int v4i __attribute__((vector_size(16)));
typedef v4i __attribute__((address_space(1))) gv4i;   // global
typedef v4i __attribute__((address_space(3))) lv4i;   // LDS
#endif

__device__ __forceinline__ void async_wait_all() {
#if defined(WAIT_ASYNC_BUILTIN)
  __builtin_amdgcn_s_wait_asynccnt(0);
#elif defined(ASYNC_LDS)
  asm volatile("s_wait_asynccnt 0" ::: "memory");
#endif
}

extern "C" __global__ void __launch_bounds__(NTHREADS)
curv_stencil(const float* __restrict__ phi, float* __restrict__ partial)
{
  __shared__ float tile[LROWS][W];                    // 64 KB
  const int tid = threadIdx.x;
  const int rb  = blockIdx.x;                        // 0..72
  const int img = blockIdx.y;                        // 0..31
  const int gr0 = 1 + rb * ROWS;                     // first interior row
  const float* base = phi + (size_t)img * W * W;
  const float* src0 = base + (size_t)(gr0 - 1) * W;  // first halo row

  // ---- global -> LDS: 16 rows x 4KB; each lane copies 16B per row ----
#if defined(ASYNC_LDS)
  {
    const char* g = (const char*)src0 + tid * 16;
    char*       l = (char*)(&tile[0][0]) + tid * 16;
#pragma unroll
    for (int r = 0; r < LROWS; ++r) {
      __builtin_amdgcn_global_load_async_to_lds_b128(
          (gv4i*)(g + (size_t)r * (W * 4)),
          (lv4i*)(l + r * (W * 4)),
          0, 0);
    }
  }
  async_wait_all();
#else
  {
#pragma unroll
    for (int r = 0; r < LROWS; ++r) {
      float4 v = *(const float4*)((const char*)src0 + (size_t)r * (W * 4) + tid * 16);
      *(float4*)((char*)(&tile[r][0]) + tid * 16) = v;
    }
  }
#endif
  __syncthreads();

  // ---- stencil with rolling 3x3 register window ----
  const float inv2h  = 1.0f / (2.0f * VOXW);
  const float inv4h2 = 1.0f / (4.0f * VOXW * VOXW);
  float accN = 0.0f, accD = 0.0f;

  for (int c = tid + 1; c < W - 1; c += NTHREADS) {
    float a0 = tile[0][c - 1], a1 = tile[0][c], a2 = tile[0][c + 1];  // row above
    float b0 = tile[1][c - 1], b1 = tile[1][c], b2 = tile[1][c + 1];  // center row
#pragma unroll
    for (int j = 0; j < ROWS; ++j) {
      const float d0 = tile[j + 2][c - 1];
      const float d1 = tile[j + 2][c];
      const float d2 = tile[j + 2][c + 1];

      const float ddx = (b2 - b0) * inv2h;          // right - left
      const float ddy = (d1 - a1) * inv2h;          // down  - up
      const float dxy = (d2 - d0 - a2 + a0) * inv4h2;
      const float dm  = ddx - ddy;
      const float num = dxy * dm * dm;              // dxy*(ddx^2 - 2 ddx ddy + ddy^2)
      const float s   = ddx * ddx + ddy * ddy;
      const float den = s * sqrtf(s);               // s^1.5
      const float curv = num / (den + EPS_F);
      const float cr   = fmaxf(curv * curv - 1.0f, 0.0f); // thresh==1 in f32

      // zero-crossing: any of 4 sign-change products negative
      const float zm = fminf(fminf(b1 * b2, b1 * b0), fminf(b1 * d1, b1 * a1));
      if (zm < 0.0f) { accN += cr; accD += 1.0f; }

      a0 = b0; a1 = b1; a2 = b2;
      b0 = d0; b1 = d1; b2 = d2;
    }
  }

  // ---- wave32 shuffle reduce, then cross-wave via (reused) LDS ----
#pragma unroll
  for (int m = 16; m >= 1; m >>= 1) {
    accN += __shfl_xor(accN, m, 32);
    accD += __shfl_xor(accD, m, 32);
  }
  __syncthreads();                                   // all tile reads done
  const int wid = tid >> 5, lane = tid & 31;
  if (lane == 0) { tile[0][wid] = accN; tile[0][8 + wid] = accD; }
  __syncthreads();
  if (tid == 0) {
    float n = 0.0f, d = 0.0f;
#pragma unroll
    for (int w = 0; w < NTHREADS / 32; ++w) { n += tile[0][w]; d += tile[0][8 + w]; }
    const int bflat = blockIdx.y * gridDim.x + blockIdx.x;
    partial[2 * bflat]     = n;
    partial[2 * bflat + 1] = d;
  }
}

extern "C" __global__ void __launch_bounds__(NTHREADS)
curv_finalize(const float* __restrict__ partial, int nblk, float* __restrict__ out)
{
  __shared__ float redN[NTHREADS / 32], redD[NTHREADS / 32];
  float n = 0.0f, d = 0.0f;
  for (int i = threadIdx.x; i < nblk; i += NTHREADS) {
    n += partial[2 * i];
    d += partial[2 * i + 1];
  }
#pragma unroll
  for (int m = 16; m >= 1; m >>= 1) {
    n += __shfl_xor(n, m, 32);
    d += __shfl_xor(d, m, 32);
  }
  const int wid = threadIdx.x >> 5, lane = threadIdx.x & 31;
  if (lane == 0) { redN[wid] = n; redD[wid] = d; }
  __syncthreads();
  if (threadIdx.x == 0) {
    float N = 0.0f, D = 0.0f;
#pragma unroll
    for (int w = 0; w < NTHREADS / 32; ++w) { N += redN[w]; D += redD[w]; }
    out[0] = N / (D + EPS_F);
  }
}

extern "C" void kernel_launch(void* const* d_in, const int* in_sizes, int n_in,
                              void* d_out, int out_size, void* d_ws, size_t ws_size,
                              hipStream_t stream)
{
  (void)in_sizes; (void)n_in; (void)out_size; (void)ws_size;
  const float* phi     = (const float*)d_in[0];
  float*       out     = (float*)d_out;
  float*       partial = (float*)d_ws;   // 2 floats per block = 2336*2*4 B ≈ 18.7 KB

  dim3 grid(RBLK, NIMG, 1);
  curv_stencil<<<grid, NTHREADS, 0, stream>>>(phi, partial);
  curv_finalize<<<1, NTHREADS, 0, stream>>>(partial, RBLK * NIMG, out);
}